// LIFLayer_73169062855155
// MI455X (gfx1250) — compile-verified
//
#include <hip/hip_runtime.h>
#include <stdint.h>

// ---------------------------------------------------------------------------
// LIF scan over time, B*F independent neurons, L=1024 sequential steps.
// Pure HBM-streaming problem: 402 MB traffic -> ~17us floor @ 23.3 TB/s,
// ~0.2 GFLOP of VALU work. No matmul structure (nonlinear threshold/reset),
// so WMMA is inapplicable; the CDNA5 win is the async-DMA pipeline:
//   global_load_async_to_lds_b128 (th:TH_LOAD_NT), prefetch distance 2
//   (triple-buffered LDS), s_wait_asynccnt graduated waits, ds_load_b128
//   compute reads, non-temporal b128 output stores.
// Depth-2 prefetch keeps 2 tiles (32KB) per block in flight; with ~5 blocks
// co-resident per WGP (LDS-limited) this maximizes outstanding bytes, which
// is what 23.3 TB/s demands.
// ---------------------------------------------------------------------------

typedef __attribute__((ext_vector_type(4))) float v4f;

#define LIF_THR  0.25f
#define LIF_BETA 10.0f
// alpha = exp(-DT/TAU) = exp(-1/3)
#define LIF_ALPHA 0.7165313105737893f

constexpr int L_STEPS = 1024;   // timesteps (known from reference shapes)
constexpr int TB      = 128;    // threads per block == neurons per block
constexpr int TT      = 32;     // timesteps per tile
constexpr int STRIDE  = 36;     // padded floats per LDS row (144B: 16B aligned)
constexpr int NTILES  = L_STEPS / TT;          // 32
constexpr int NLOADS  = (TB * TT / 4) / TB;    // 8 async b128 instrs per tile
constexpr int NBUF    = 3;                     // prefetch distance 2

__global__ __launch_bounds__(TB)
void lif_scan_kernel(const float* __restrict__ I,
                     float* __restrict__ Z,
                     float* __restrict__ S)
{
    __shared__ __align__(16) float sh[NBUF][TB * STRIDE];

    const int    tid   = threadIdx.x;
    const size_t nbase = (size_t)blockIdx.x * TB;
    const float* gin   = I + nbase * L_STEPS;

    // Low 32 bits of a generic pointer to LDS == LDS byte offset (the
    // generic->AS(3) cast on amdgcn is a truncation).
    const uint32_t lds_base = (uint32_t)(uintptr_t)(&sh[0][0]);

    // Issue one tile's worth of async global->LDS b128 copies.
    // Lane-op q = j*TB + tid; 8 lanes cover one 128B row segment, so every
    // async instruction moves 4 fully-used 128B lines. Input is read once:
    // mark non-temporal.
    auto issue_tile = [&](int tile, int buf) {
        const float*   gt    = gin + tile * TT;
        const uint32_t lbase = lds_base + (uint32_t)buf * (TB * STRIDE * 4);
#pragma unroll
        for (int j = 0; j < NLOADS; ++j) {
            const int q   = j * TB + tid;
            const int row = q >> 3;   // neuron row inside the block
            const int seg = q & 7;    // 16B segment inside the 128B row
            const float*   ga = gt + (size_t)row * L_STEPS + seg * 4;
            const uint32_t la = lbase + (uint32_t)(row * STRIDE + seg * 4) * 4u;
            asm volatile("global_load_async_to_lds_b128 %0, %1, off th:TH_LOAD_NT"
                         :: "v"(la), "v"(ga)
                         : "memory");
        }
    };

    // Prologue: fill the pipeline two tiles deep.
    issue_tile(0, 0);
    issue_tile(1, 1);

    float v = 0.0f;
    float* zrow = Z + (nbase + tid) * L_STEPS;
    float* srow = S + (nbase + tid) * L_STEPS;

    for (int t = 0; t < NTILES; ++t) {
        const int buf = t % NBUF;

        // Keep 2 tiles in flight; async loads complete in order, so
        // "<= 8*pending_after" guarantees tile t has fully landed.
        if (t + 2 < NTILES) {
            issue_tile(t + 2, (t + 2) % NBUF);
            asm volatile("s_wait_asynccnt 0x10" ::: "memory");
        } else if (t + 1 < NTILES) {
            asm volatile("s_wait_asynccnt 0x8" ::: "memory");
        } else {
            asm volatile("s_wait_asynccnt 0x0" ::: "memory");
        }
        __syncthreads();   // tile t visible to the whole workgroup

        const float* rowp = &sh[buf][tid * STRIDE];
        float* zt = zrow + t * TT;
        float* st = srow + t * TT;

#pragma unroll
        for (int kk = 0; kk < TT / 4; ++kk) {
            v4f iv = *(const v4f*)(rowp + kk * 4);   // ds_load_b128
            v4f zv, sv;
#pragma unroll
            for (int e = 0; e < 4; ++e) {
                const float vpre  = LIF_ALPHA * v + (1.0f - LIF_ALPHA) * iv[e];
                const bool  spike = (vpre >= LIF_THR);
                zv[e] = LIF_BETA * (vpre - LIF_THR);
                sv[e] = spike ? 1.0f : 0.0f;
                v     = spike ? 0.0f : vpre;
            }
            // Streamed outputs: each thread emits one full 128B line per
            // tile per tensor in back-to-back NT b128 stores (L2 combines).
            __builtin_nontemporal_store(zv, (v4f*)(zt + kk * 4));
            __builtin_nontemporal_store(sv, (v4f*)(st + kk * 4));
        }
        __syncthreads();   // all waves done reading buf before it is reused
    }
}

extern "C" void kernel_launch(void* const* d_in, const int* in_sizes, int n_in,
                              void* d_out, int out_size, void* d_ws, size_t ws_size,
                              hipStream_t stream)
{
    const float* I = (const float*)d_in[0];
    float*       out = (float*)d_out;

    const size_t total = (size_t)in_sizes[0];     // B*F*L = 33554432
    const size_t N     = total / L_STEPS;         // B*F neurons = 32768

    float* Z = out;                   // z history, (B,F,L) flat
    float* S = out + N * L_STEPS;     // s history, (B,F,L) flat

    const int blocks = (int)(N / TB); // 256
    lif_scan_kernel<<<blocks, TB, 0, stream>>>(I, Z, S);
}